// GCN_60533269069867
// MI455X (gfx1250) — compile-verified
//
#include <hip/hip_runtime.h>
#include <hip/hip_bf16.h>

typedef __attribute__((ext_vector_type(2))) float v2f;
typedef __attribute__((ext_vector_type(8))) float v8f;

__device__ __forceinline__ v8f wmma_f32_k4(v2f a, v2f b, v8f c) {
    // V_WMMA_F32_16X16X4_F32 : D = A(16x4) * B(4x16) + C(16x16), all fp32
    return __builtin_amdgcn_wmma_f32_16x16x4_f32(
        /*neg_a=*/false, a, /*neg_b=*/false, b,
        /*c_mod=*/(short)0, c, /*reuse_a=*/false, /*reuse_b=*/false);
}

__device__ __forceinline__ void atomAddF(float* p, float v) {
    __hip_atomic_fetch_add(p, v, __ATOMIC_RELAXED, __HIP_MEMORY_SCOPE_AGENT);
}

// ---------------- degree / norm ----------------
__global__ void k_init_deg(float* deg, int N) {
    int i = blockIdx.x * blockDim.x + threadIdx.x;
    if (i < N) deg[i] = 1.0f;                 // self-loop contributes 1
}
__global__ void k_deg_accum(const int* __restrict__ dst, float* deg, int E) {
    int e = blockIdx.x * blockDim.x + threadIdx.x;
    if (e < E) atomAddF(&deg[dst[e]], 1.0f);
}
__global__ void k_rsqrt(float* deg, int N) {
    int i = blockIdx.x * blockDim.x + threadIdx.x;
    if (i < N) deg[i] = rsqrtf(deg[i]);       // deg >= 1 always
}

// ---------------- zero fill ----------------
__global__ void k_zero(float* p, long long n) {
    long long i = (long long)blockIdx.x * blockDim.x + threadIdx.x;
    if (i < n) p[i] = 0.0f;
}

// ---------------- GEMM1: xw = x(N x 300) @ W1(300 x 128), WMMA f32 ----------------
// block = 256 threads = 8 waves; wave w computes the 16x16 tile (mt, n = 16*w)
__global__ void k_gemm1(const float* __restrict__ x, const float* __restrict__ W1,
                        float* __restrict__ xw) {
    const int lane = threadIdx.x & 31;
    const int wave = threadIdx.x >> 5;        // 0..7 -> N tile
    const int half = lane >> 4;               // 0: K={0,1}/M rows, 1: K={2,3}
    const int rlo  = lane & 15;
    const int mt   = blockIdx.x;              // M tile (N/16 tiles)
    const int row  = (mt << 4) + rlo;         // A row for this lane
    const int ncol = (wave << 4) + rlo;       // B/C column for this lane

    v8f acc = {};
    const float* xrow = x + (size_t)row * 300 + (half << 1);
    for (int k = 0; k < 300; k += 4) {
        float2 av = *(const float2*)(xrow + k);          // A[row][k+2*half .. +1]
        v2f a; a.x = av.x; a.y = av.y;
        const int kr = k + (half << 1);
        v2f b;
        b.x = W1[(size_t)kr * 128 + ncol];               // B[kr  ][ncol]
        b.y = W1[(size_t)(kr + 1) * 128 + ncol];         // B[kr+1][ncol]
        acc = wmma_f32_k4(a, b, acc);
    }
    // C layout: VGPR r -> M = r + 8*half, N = ncol
    const int m0 = (mt << 4) + (half << 3);
    #pragma unroll
    for (int r = 0; r < 8; ++r)
        xw[(size_t)(m0 + r) * 128 + ncol] = acc[r];
}

// ---------------- aggregation layer 1 (128 feats, 32 lanes x float4/edge) ----------------
__global__ void k_agg1(const float* __restrict__ xw, const int* __restrict__ src,
                       const int* __restrict__ dst, const float* __restrict__ dinv,
                       float* __restrict__ agg, int E) {
    long long gid = (long long)blockIdx.x * blockDim.x + threadIdx.x;
    int e = (int)(gid >> 5);
    int f = ((int)gid & 31) << 2;
    if (e >= E) return;
    int s = src[e], d = dst[e];
    float nrm = dinv[s] * dinv[d];
    float4 v = *(const float4*)(xw + (size_t)s * 128 + f);
    float* o = agg + (size_t)d * 128 + f;
    atomAddF(o + 0, v.x * nrm);
    atomAddF(o + 1, v.y * nrm);
    atomAddF(o + 2, v.z * nrm);
    atomAddF(o + 3, v.w * nrm);
}

// ---------------- h = relu(agg + xw * dinv^2 + b1), in place on agg ----------------
__global__ void k_bias_relu(float* __restrict__ h, const float* __restrict__ xw,
                            const float* __restrict__ dinv, const float* __restrict__ b1,
                            int N) {
    long long gid = (long long)blockIdx.x * blockDim.x + threadIdx.x;
    if (gid >= (long long)N * 128) return;
    int node = (int)(gid >> 7);
    int f = (int)gid & 127;
    float di = dinv[node];
    float v = h[gid] + xw[gid] * di * di + b1[f];
    h[gid] = v > 0.0f ? v : 0.0f;
}

// ---------------- GEMM2: hw = h(N x 128) @ W2(128 x 10 -> padded 16), WMMA f32 ----------------
__global__ void k_gemm2(const float* __restrict__ h, const float* __restrict__ W2,
                        float* __restrict__ hw16, int N) {
    __shared__ float w2s[128 * 16];                       // W2 zero-padded to 16 cols
    for (int i = threadIdx.x; i < 128 * 16; i += 256) {
        int k = i >> 4, c = i & 15;
        w2s[i] = (c < 10) ? W2[k * 10 + c] : 0.0f;
    }
    __syncthreads();

    const int lane = threadIdx.x & 31;
    const int wave = threadIdx.x >> 5;
    const int half = lane >> 4;
    const int rlo  = lane & 15;
    const int mt   = blockIdx.x * 8 + wave;               // M tile
    if ((mt << 4) < N) {                                  // wave-uniform tail guard
        v8f acc = {};
        const float* hrow = h + (size_t)((mt << 4) + rlo) * 128 + (half << 1);
        for (int k = 0; k < 128; k += 4) {
            float2 av = *(const float2*)(hrow + k);
            v2f a; a.x = av.x; a.y = av.y;
            const int kr = k + (half << 1);
            v2f b;
            b.x = w2s[kr * 16 + rlo];
            b.y = w2s[(kr + 1) * 16 + rlo];
            acc = wmma_f32_k4(a, b, acc);
        }
        const int m0 = (mt << 4) + (half << 3);
        #pragma unroll
        for (int r = 0; r < 8; ++r)
            hw16[(size_t)(m0 + r) * 16 + rlo] = acc[r];
    }
}

// ---------------- aggregation layer 2 (10 feats, 16 threads/edge) ----------------
__global__ void k_agg2(const float* __restrict__ hw16, const int* __restrict__ src,
                       const int* __restrict__ dst, const float* __restrict__ dinv,
                       float* __restrict__ out, int E) {
    long long gid = (long long)blockIdx.x * blockDim.x + threadIdx.x;
    int e = (int)(gid >> 4);
    int f = (int)gid & 15;
    if (e >= E || f >= 10) return;
    int s = src[e], d = dst[e];
    float nrm = dinv[s] * dinv[d];
    atomAddF(out + (size_t)d * 10 + f, hw16[(size_t)s * 16 + f] * nrm);
}

// ---------------- out += hw * dinv^2 + b2 (self-loop + bias) ----------------
__global__ void k_final(const float* __restrict__ hw16, const float* __restrict__ dinv,
                        const float* __restrict__ b2, float* __restrict__ out, int N) {
    long long gid = (long long)blockIdx.x * blockDim.x + threadIdx.x;
    if (gid >= (long long)N * 10) return;
    int node = (int)(gid / 10);
    int f = (int)(gid % 10);
    float di = dinv[node];
    out[gid] += hw16[(size_t)node * 16 + f] * di * di + b2[f];
}

extern "C" void kernel_launch(void* const* d_in, const int* in_sizes, int n_in,
                              void* d_out, int out_size, void* d_ws, size_t ws_size,
                              hipStream_t stream) {
    const float* x  = (const float*)d_in[0];
    const int*   ei = (const int*)d_in[1];
    const float* W1 = (const float*)d_in[2];
    const float* b1 = (const float*)d_in[3];
    const float* W2 = (const float*)d_in[4];
    const float* b2 = (const float*)d_in[5];
    float* out = (float*)d_out;

    const int N = in_sizes[0] / 300;          // 100000
    const int E = in_sizes[1] / 2;            // 1600000
    const int* src = ei;
    const int* dst = ei + E;

    // workspace carve-up (64-float aligned)
    size_t Np = ((size_t)N + 63) & ~(size_t)63;
    float* dinv = (float*)d_ws;               // N
    float* xw   = dinv + Np;                  // N x 128
    float* h    = xw + Np * 128;              // N x 128 (agg1, then relu'd h in place)
    float* hw   = h  + Np * 128;              // N x 16 (padded layer-2 xw)

    const int T = 256;
    const long long nf1 = (long long)N * 128;

    k_init_deg <<<(N + T - 1) / T, T, 0, stream>>>(dinv, N);
    k_deg_accum<<<(E + T - 1) / T, T, 0, stream>>>(dst, dinv, E);
    k_rsqrt    <<<(N + T - 1) / T, T, 0, stream>>>(dinv, N);

    k_gemm1    <<<N / 16, T, 0, stream>>>(x, W1, xw);

    k_zero     <<<(unsigned)((nf1 + T - 1) / T), T, 0, stream>>>(h, nf1);
    k_agg1     <<<(unsigned)(((long long)E * 32 + T - 1) / T), T, 0, stream>>>(
                    xw, src, dst, dinv, h, E);
    k_bias_relu<<<(unsigned)((nf1 + T - 1) / T), T, 0, stream>>>(h, xw, dinv, b1, N);

    k_gemm2    <<<(N / 16 + 7) / 8, T, 0, stream>>>(h, W2, hw, N);

    const long long no = (long long)N * 10;
    k_zero     <<<(unsigned)((no + T - 1) / T), T, 0, stream>>>(out, no);
    k_agg2     <<<(unsigned)(((long long)E * 16 + T - 1) / T), T, 0, stream>>>(
                    hw, src, dst, dinv, out, E);
    k_final    <<<(unsigned)((no + T - 1) / T), T, 0, stream>>>(hw, dinv, b2, out, N);
}